// EmbeddingLayer_64450279244312
// MI455X (gfx1250) — compile-verified
//
#include <hip/hip_runtime.h>
#include <math.h>

#define NUM_RADIAL 6
#define EMB 128
#define N_NODES 50000
#define N_EDGES 800000
#define K3 384          // 3*EMB
#define NODE_TILES (N_NODES / 16)
#define EDGE_TILES (N_EDGES / 16)

typedef __attribute__((ext_vector_type(2))) float v2f;
typedef __attribute__((ext_vector_type(8))) float v8f;

__device__ __forceinline__ v8f wmma4(v2f a, v2f b, v8f c) {
  // V_WMMA_F32_16X16X4_F32: D = A(16x4) x B(4x16) + C(16x16), all f32.
  return __builtin_amdgcn_wmma_f32_16x16x4_f32(
      /*neg_a=*/false, a, /*neg_b=*/false, b,
      /*c_mod=*/(short)0, c, /*reuse_a=*/false, /*reuse_b=*/false);
}

__device__ __forceinline__ float silu_f(float x) {
  return x / (1.0f + __expf(-x));
}

__device__ __forceinline__ float envelope_f(float x) {
  // p = ENV_EXP + 1 = 6:  1/x - 28 x^5 + 48 x^6 - 21 x^7
  float x2 = x * x;
  float x4 = x2 * x2;
  float x5 = x4 * x;
  return 1.0f / x + x5 * (-28.0f + x * (48.0f - 21.0f * x));
}

// ---------------------------------------------------------------------------
// Node kernel: h = emb_table[Z]  (written to output)
// optionally:  Hs = h @ W1^T,  Hd = h @ W2^T + b_dense   (to workspace)
// One block = 16 nodes; 8 waves each own a 16-wide N-tile of the 128 outputs.
// ---------------------------------------------------------------------------
__global__ __launch_bounds__(256) void dimenet_node_kernel(
    const int* __restrict__ Z,
    const float* __restrict__ emb_table,
    const float* __restrict__ W_dense,   // [128][384] row-major
    const float* __restrict__ b_dense,   // [128]
    float* __restrict__ h_out,           // [N_NODES][128]
    float* __restrict__ Hs,              // [N_NODES][128] (workspace) or null
    float* __restrict__ Hd,              // [N_NODES][128] (workspace) or null
    int do_pre) {
  __shared__ float htile[16][132];

  const int t = threadIdx.x;
  const int lane = t & 31;
  const int wave = t >> 5;
  const int ln = lane & 15;
  const int hi = lane >> 4;
  const int n0 = blockIdx.x * 16;

  // Stage 16x128 h tile into LDS and write h output (float4-coalesced).
  {
    const int e = t >> 4;
    const int j0 = (t & 15) << 3;
    const int n = n0 + e;
    const int z = Z[n];
    float4 a = *(const float4*)(emb_table + (size_t)z * EMB + j0);
    float4 b = *(const float4*)(emb_table + (size_t)z * EMB + j0 + 4);
    *(float4*)(&htile[e][j0])     = a;
    *(float4*)(&htile[e][j0 + 4]) = b;
    *(float4*)(h_out + (size_t)n * EMB + j0)     = a;
    *(float4*)(h_out + (size_t)n * EMB + j0 + 4) = b;
  }
  __syncthreads();
  if (!do_pre) return;

  const int jb = wave * 16;
  v8f cS = {};
  v8f cD;
  {
    float bv = b_dense[jb + ln];   // bias depends on N only
#pragma unroll
    for (int v = 0; v < 8; ++v) cD[v] = bv;
  }

  const float* Wrow = W_dense + (size_t)(jb + ln) * K3;   // row j of W_dense
#pragma unroll 4
  for (int kb = 0; kb < EMB; kb += 4) {
    v2f a  = *(const v2f*)(&htile[ln][kb + 2 * hi]);        // A[M=ln][K]
    v2f b1 = *(const v2f*)(Wrow + kb + 2 * hi);             // W1: B[K][N=ln]
    v2f b2 = *(const v2f*)(Wrow + EMB + kb + 2 * hi);       // W2
    cS = wmma4(a, b1, cS);
    cD = wmma4(a, b2, cD);
  }

#pragma unroll
  for (int v = 0; v < 8; ++v) {
    const int mr = v + 8 * hi;
    Hs[(size_t)(n0 + mr) * EMB + jb + ln] = cS[v];
    Hd[(size_t)(n0 + mr) * EMB + jb + ln] = cD[v];
  }
}

// ---------------------------------------------------------------------------
// Edge kernel (precomputed path):
//   m[e] = silu( Hs[src[e]] + Hd[dst[e]] + silu(rbf[e]@W_rbf^T + b_rbf) @ W3^T )
//   rbf_env[e] = envelope(d/c) * sin(n*pi*d/c)/(d/c)
// W3 staged in LDS once per block; grid-stride over 16-edge tiles.
// ---------------------------------------------------------------------------
__global__ __launch_bounds__(256) void dimenet_edge_pre_kernel(
    const int* __restrict__ src, const int* __restrict__ dst,
    const float* __restrict__ rbf, const float* __restrict__ dists,
    const float* __restrict__ W_rbf, const float* __restrict__ b_rbf,
    const float* __restrict__ W_dense,
    const float* __restrict__ Hs, const float* __restrict__ Hd,
    float* __restrict__ m_out, float* __restrict__ env_out) {
  __shared__ float w3[128][132];          // W3[j][k] = W_dense[j][256+k]
  __shared__ float atile[16][132];        // rbf_proj tile (A matrix)
  __shared__ float acc[16][132];          // Hs[src]+Hd[dst] tile (C init)
  __shared__ float wrbf_s[EMB][8];
  __shared__ float brbf_s[EMB];
  __shared__ float rtile[16][NUM_RADIAL];

  const int t = threadIdx.x;
  const int lane = t & 31;
  const int wave = t >> 5;
  const int ln = lane & 15;
  const int hi = lane >> 4;
  const int jb = wave * 16;
  const int e_loc = t >> 4;
  const int j0 = (t & 15) << 3;

  // Stage W3 (128x128) once per block, padded stride for conflict-free frags.
  for (int i = t; i < 128 * 32; i += 256) {
    const int j = i >> 5, k4 = (i & 31) << 2;
    float4 v = *(const float4*)(W_dense + (size_t)j * K3 + 256 + k4);
    *(float4*)(&w3[j][k4]) = v;
  }
  if (t < EMB) {
#pragma unroll
    for (int r = 0; r < NUM_RADIAL; ++r) wrbf_s[t][r] = W_rbf[t * NUM_RADIAL + r];
    brbf_s[t] = b_rbf[t];
  }
  __syncthreads();

  for (int tile = blockIdx.x; tile < EDGE_TILES; tile += gridDim.x) {
    const int e0 = tile * 16;

    // ---- phase 1: rbf staging + rbf_env + Hs/Hd gather-accumulate ----
    if (t < 16 * NUM_RADIAL) {
      const int e = t / NUM_RADIAL, r = t - e * NUM_RADIAL;
      rtile[e][r] = rbf[(size_t)(e0 + e) * NUM_RADIAL + r];
      const float x = dists[e0 + e] * 0.2f;                 // d / CUTOFF
      const float freq = 3.14159265358979323846f * (float)(r + 1);
      env_out[(size_t)(e0 + e) * NUM_RADIAL + r] =
          envelope_f(x) * (sinf(x * freq) / x);
    }
    {
      const int s = src[e0 + e_loc];
      const int dd = dst[e0 + e_loc];
      float4 a0 = *(const float4*)(Hs + (size_t)s * EMB + j0);
      float4 a1 = *(const float4*)(Hs + (size_t)s * EMB + j0 + 4);
      float4 b0 = *(const float4*)(Hd + (size_t)dd * EMB + j0);
      float4 b1 = *(const float4*)(Hd + (size_t)dd * EMB + j0 + 4);
      float4 r0 = make_float4(a0.x + b0.x, a0.y + b0.y, a0.z + b0.z, a0.w + b0.w);
      float4 r1 = make_float4(a1.x + b1.x, a1.y + b1.y, a1.z + b1.z, a1.w + b1.w);
      *(float4*)(&acc[e_loc][j0])     = r0;
      *(float4*)(&acc[e_loc][j0 + 4]) = r1;
    }
    __syncthreads();

    // ---- phase 2: rbf_proj = silu(rbf @ W_rbf^T + b_rbf) ----
#pragma unroll
    for (int i = 0; i < 8; ++i) {
      const int k = j0 + i;
      float s = brbf_s[k];
#pragma unroll
      for (int r = 0; r < NUM_RADIAL; ++r) s += rtile[e_loc][r] * wrbf_s[k][r];
      atile[e_loc][k] = silu_f(s);
    }
    __syncthreads();

    // ---- phase 3: C = acc;  C += atile(16x128) x W3^T(128x16);  silu ----
    v8f c;
#pragma unroll
    for (int v = 0; v < 8; ++v) c[v] = acc[v + 8 * hi][jb + ln];
#pragma unroll 8
    for (int kb = 0; kb < EMB; kb += 4) {
      v2f a = *(const v2f*)(&atile[ln][kb + 2 * hi]);
      v2f b = *(const v2f*)(&w3[jb + ln][kb + 2 * hi]);
      c = wmma4(a, b, c);
    }
#pragma unroll
    for (int v = 0; v < 8; ++v) {
      const int mr = v + 8 * hi;
      m_out[(size_t)(e0 + mr) * EMB + jb + ln] = silu_f(c[v]);
    }
    __syncthreads();
  }
}

// ---------------------------------------------------------------------------
// Fallback edge kernel (no workspace): direct K=384 GEMM.
//   A tiles: h[src], h[dst], rbf_proj (each 16x128 in LDS); B from global.
// ---------------------------------------------------------------------------
__global__ __launch_bounds__(256) void dimenet_edge_direct_kernel(
    const int* __restrict__ src, const int* __restrict__ dst,
    const float* __restrict__ rbf, const float* __restrict__ dists,
    const float* __restrict__ W_rbf, const float* __restrict__ b_rbf,
    const float* __restrict__ W_dense, const float* __restrict__ b_dense,
    const float* __restrict__ h,        // node embeddings (already gathered)
    float* __restrict__ m_out, float* __restrict__ env_out) {
  __shared__ float hs_t[16][132];
  __shared__ float hd_t[16][132];
  __shared__ float atile[16][132];
  __shared__ float wrbf_s[EMB][8];
  __shared__ float brbf_s[EMB];
  __shared__ float rtile[16][NUM_RADIAL];

  const int t = threadIdx.x;
  const int lane = t & 31;
  const int wave = t >> 5;
  const int ln = lane & 15;
  const int hi = lane >> 4;
  const int jb = wave * 16;
  const int e_loc = t >> 4;
  const int j0 = (t & 15) << 3;

  if (t < EMB) {
#pragma unroll
    for (int r = 0; r < NUM_RADIAL; ++r) wrbf_s[t][r] = W_rbf[t * NUM_RADIAL + r];
    brbf_s[t] = b_rbf[t];
  }
  __syncthreads();

  for (int tile = blockIdx.x; tile < EDGE_TILES; tile += gridDim.x) {
    const int e0 = tile * 16;
    if (t < 16 * NUM_RADIAL) {
      const int e = t / NUM_RADIAL, r = t - e * NUM_RADIAL;
      rtile[e][r] = rbf[(size_t)(e0 + e) * NUM_RADIAL + r];
      const float x = dists[e0 + e] * 0.2f;
      const float freq = 3.14159265358979323846f * (float)(r + 1);
      env_out[(size_t)(e0 + e) * NUM_RADIAL + r] =
          envelope_f(x) * (sinf(x * freq) / x);
    }
    {
      const int s = src[e0 + e_loc];
      const int dd = dst[e0 + e_loc];
      *(float4*)(&hs_t[e_loc][j0])     = *(const float4*)(h + (size_t)s * EMB + j0);
      *(float4*)(&hs_t[e_loc][j0 + 4]) = *(const float4*)(h + (size_t)s * EMB + j0 + 4);
      *(float4*)(&hd_t[e_loc][j0])     = *(const float4*)(h + (size_t)dd * EMB + j0);
      *(float4*)(&hd_t[e_loc][j0 + 4]) = *(const float4*)(h + (size_t)dd * EMB + j0 + 4);
    }
    __syncthreads();

#pragma unroll
    for (int i = 0; i < 8; ++i) {
      const int k = j0 + i;
      float s = brbf_s[k];
#pragma unroll
      for (int r = 0; r < NUM_RADIAL; ++r) s += rtile[e_loc][r] * wrbf_s[k][r];
      atile[e_loc][k] = silu_f(s);
    }
    __syncthreads();

    v8f c;
    {
      float bv = b_dense[jb + ln];
#pragma unroll
      for (int v = 0; v < 8; ++v) c[v] = bv;
    }
    const float* Wrow = W_dense + (size_t)(jb + ln) * K3;
#pragma unroll 4
    for (int kb = 0; kb < EMB; kb += 4) {
      v2f a = *(const v2f*)(&hs_t[ln][kb + 2 * hi]);
      v2f b = *(const v2f*)(Wrow + kb + 2 * hi);
      c = wmma4(a, b, c);
    }
#pragma unroll 4
    for (int kb = 0; kb < EMB; kb += 4) {
      v2f a = *(const v2f*)(&hd_t[ln][kb + 2 * hi]);
      v2f b = *(const v2f*)(Wrow + EMB + kb + 2 * hi);
      c = wmma4(a, b, c);
    }
#pragma unroll 4
    for (int kb = 0; kb < EMB; kb += 4) {
      v2f a = *(const v2f*)(&atile[ln][kb + 2 * hi]);
      v2f b = *(const v2f*)(Wrow + 2 * EMB + kb + 2 * hi);
      c = wmma4(a, b, c);
    }
#pragma unroll
    for (int v = 0; v < 8; ++v) {
      const int mr = v + 8 * hi;
      m_out[(size_t)(e0 + mr) * EMB + jb + ln] = silu_f(c[v]);
    }
    __syncthreads();
  }
}

extern "C" void kernel_launch(void* const* d_in, const int* in_sizes, int n_in,
                              void* d_out, int out_size, void* d_ws, size_t ws_size,
                              hipStream_t stream) {
  (void)in_sizes; (void)n_in; (void)out_size;
  const int*   Z       = (const int*)d_in[0];
  const int*   src     = (const int*)d_in[1];
  const int*   dst     = (const int*)d_in[2];
  const float* rbf     = (const float*)d_in[3];
  const float* dists   = (const float*)d_in[4];
  const float* emb     = (const float*)d_in[5];
  const float* W_rbf   = (const float*)d_in[6];
  const float* b_rbf   = (const float*)d_in[7];
  const float* W_dense = (const float*)d_in[8];
  const float* b_dense = (const float*)d_in[9];

  float* out     = (float*)d_out;
  float* h_out   = out;                                    // [50000*128]
  float* m_out   = out + (size_t)N_NODES * EMB;            // [800000*128]
  float* env_out = m_out + (size_t)N_EDGES * EMB;          // [800000*6]

  const size_t need = (size_t)2 * N_NODES * EMB * sizeof(float);
  const bool pre = (ws_size >= need);
  float* Hs = (float*)d_ws;
  float* Hd = Hs + (size_t)N_NODES * EMB;

  dimenet_node_kernel<<<NODE_TILES, 256, 0, stream>>>(
      Z, emb, W_dense, b_dense, h_out, pre ? Hs : nullptr, pre ? Hd : nullptr,
      pre ? 1 : 0);

  const int grid = 4096;
  if (pre) {
    dimenet_edge_pre_kernel<<<grid, 256, 0, stream>>>(
        src, dst, rbf, dists, W_rbf, b_rbf, W_dense, Hs, Hd, m_out, env_out);
  } else {
    dimenet_edge_direct_kernel<<<grid, 256, 0, stream>>>(
        src, dst, rbf, dists, W_rbf, b_rbf, W_dense, b_dense, h_out, m_out,
        env_out);
  }
}